// ToiPool_9646496547022
// MI455X (gfx1250) — compile-verified
//
#include <hip/hip_runtime.h>
#include <stdint.h>

#define B_ 4
#define C_ 256
#define T_ 16
#define H_ 56
#define W_ 56
#define N_ 256
#define FR 8
#define PH_ 7
#define PW_ 7
#define HW_ (H_ * W_)          // 3136
#define P_  (T_ * HW_)         // 50176 (multiple of 32)
#define CELLS (PH_ * PW_)      // 49
#define OUTSTRIDE (FR * CELLS) // 392

typedef __attribute__((ext_vector_type(4))) float v4f;

// ---------------------------------------------------------------------------
// Kernel 1: transpose [B][C][T*H*W] -> [B][T*H*W][C] so the channel dimension
// becomes contiguous (corner gathers in the pool kernel become coalesced 1KB
// vectors). Tile fill uses CDNA5 async global->LDS DMA (ASYNCcnt path) with a
// non-temporal hint: the source is read once, the destination is re-read hard
// and nearly fits the 192MB L2 — don't let the former evict the latter.
// ---------------------------------------------------------------------------
__global__ __launch_bounds__(256) void transpose_cthw_to_thwc(
    const float* __restrict__ in, float* __restrict__ out) {
  __shared__ float tile[32][33];  // +1 pad: conflict-free transposed reads
  const int b  = blockIdx.z;
  const int c0 = blockIdx.y * 32;
  const int p0 = blockIdx.x * 32;
  const int tx = threadIdx.x;     // 0..31
  const int ty = threadIdx.y;     // 0..7
  const float* src = in  + (size_t)b * C_ * P_;
  float*       dst = out + (size_t)b * P_ * C_;

#pragma unroll
  for (int i = 0; i < 4; ++i) {
    const int c = c0 + ty + i * 8;
    const int p = p0 + tx;
    // LDS byte offset = low 32 bits of the flat shared-aperture address
    unsigned ldsoff = (unsigned)(uintptr_t)(&tile[ty + i * 8][tx]);
    unsigned long long ga =
        (unsigned long long)(uintptr_t)(src + (size_t)c * P_ + (size_t)p);
    // CDNA5 async copy: global memory -> LDS, tracked by ASYNCcnt
    asm volatile("global_load_async_to_lds_b32 %0, %1, off th:TH_LOAD_NT"
                 :: "v"(ldsoff), "v"(ga)
                 : "memory");
  }
  asm volatile("s_wait_asynccnt 0" ::: "memory");
  __syncthreads();

#pragma unroll
  for (int i = 0; i < 4; ++i) {
    const int p = p0 + ty + i * 8;
    const int c = c0 + tx;
    dst[(size_t)p * C_ + (size_t)c] = tile[tx][ty + i * 8];
  }
}

// ---------------------------------------------------------------------------
// Kernel 2: ToI pooling. One 256-thread block per (roi n, output frame ft).
// Wave w handles cells s = w, w+8, ...; lanes stride channels. Transposed
// layout -> each corner load is a coalesced b128 per lane (wave covers 128
// channels per corner per chunk; 2 chunks cover C=256).
// ---------------------------------------------------------------------------
__device__ __forceinline__ size_t feat_base(int b, int t, int y, int x,
                                            int transposed) {
  if (transposed)  // [B][T][H][W][C], c added separately (+c)
    return ((size_t)((b * T_ + t) * HW_ + y * W_ + x)) * C_;
  // [B][C][T][H][W], c added separately (+c*P_)
  return ((size_t)(b * C_) * T_ + (size_t)t) * HW_ + (size_t)(y * W_ + x);
}

__global__ __launch_bounds__(256) void toi_pool(
    const float* __restrict__ feat, const float* __restrict__ rois,
    float* __restrict__ out, int transposed) {
  __shared__ float buf[CELLS * C_];  // 49*256 floats = 50176 B, layout [c][s]

  const int n  = blockIdx.x >> 3;
  const int ft = blockIdx.x & 7;
  const float* r = rois + n * 7;
  const int   b  = (int)r[0];
  const float t1 = r[1], x1 = r[2], y1 = r[3];
  const float t2 = r[4], x2 = r[5], y2 = r[6];

  // temporal center / corners for this output frame
  float tc = t1 + ((float)ft + 0.5f) * (1.0f / FR) * (t2 - t1);
  tc = fminf(fmaxf(tc, 0.0f), (float)(T_ - 1));
  const float tlo = floorf(tc);
  const float wt  = tc - tlo;
  const int   t0  = (int)tlo;
  const int   t1i = min(t0 + 1, T_ - 1);

  const int tid  = threadIdx.x;
  const int wave = tid >> 5;
  const int lane = tid & 31;

  for (int s = wave; s < CELLS; s += 8) {
    const int py = s / PW_;
    const int px = s - py * PW_;

    float yc = y1 + ((float)py + 0.5f) * (1.0f / PH_) * (y2 - y1);
    yc = fminf(fmaxf(yc, 0.0f), (float)(H_ - 1));
    const float ylo = floorf(yc);
    const float wy  = yc - ylo;
    const int   y0  = (int)ylo;
    const int   y1i = min(y0 + 1, H_ - 1);

    float xc = x1 + ((float)px + 0.5f) * (1.0f / PW_) * (x2 - x1);
    xc = fminf(fmaxf(xc, 0.0f), (float)(W_ - 1));
    const float xlo = floorf(xc);
    const float wx  = xc - xlo;
    const int   x0  = (int)xlo;
    const int   x1i = min(x0 + 1, W_ - 1);

    const float w000 = (1.0f - wt) * (1.0f - wy) * (1.0f - wx);
    const float w001 = (1.0f - wt) * (1.0f - wy) * wx;
    const float w010 = (1.0f - wt) * wy * (1.0f - wx);
    const float w011 = (1.0f - wt) * wy * wx;
    const float w100 = wt * (1.0f - wy) * (1.0f - wx);
    const float w101 = wt * (1.0f - wy) * wx;
    const float w110 = wt * wy * (1.0f - wx);
    const float w111 = wt * wy * wx;

    const size_t i000 = feat_base(b, t0,  y0,  x0,  transposed);
    const size_t i001 = feat_base(b, t0,  y0,  x1i, transposed);
    const size_t i010 = feat_base(b, t0,  y1i, x0,  transposed);
    const size_t i011 = feat_base(b, t0,  y1i, x1i, transposed);
    const size_t i100 = feat_base(b, t1i, y0,  x0,  transposed);
    const size_t i101 = feat_base(b, t1i, y0,  x1i, transposed);
    const size_t i110 = feat_base(b, t1i, y1i, x0,  transposed);
    const size_t i111 = feat_base(b, t1i, y1i, x1i, transposed);

    if (transposed) {
      // b128 path: corner bases are 1KB-aligned contiguous channel vectors
      const v4f* f000 = (const v4f*)(feat + i000);
      const v4f* f001 = (const v4f*)(feat + i001);
      const v4f* f010 = (const v4f*)(feat + i010);
      const v4f* f011 = (const v4f*)(feat + i011);
      const v4f* f100 = (const v4f*)(feat + i100);
      const v4f* f101 = (const v4f*)(feat + i101);
      const v4f* f110 = (const v4f*)(feat + i110);
      const v4f* f111 = (const v4f*)(feat + i111);
#pragma unroll
      for (int chunk = 0; chunk < C_ / 128; ++chunk) {
        const int c4 = (chunk << 5) + lane;  // float4 index; c = 4*c4
        v4f v;
        v  = w000 * f000[c4];
        v += w001 * f001[c4];
        v += w010 * f010[c4];
        v += w011 * f011[c4];
        v += w100 * f100[c4];
        v += w101 * f101[c4];
        v += w110 * f110[c4];
        v += w111 * f111[c4];
        const int cb = 4 * c4;
        buf[(cb + 0) * CELLS + s] = v.x;
        buf[(cb + 1) * CELLS + s] = v.y;
        buf[(cb + 2) * CELLS + s] = v.z;
        buf[(cb + 3) * CELLS + s] = v.w;
      }
    } else {
      // fallback (original layout): channel stride is P_ floats
#pragma unroll
      for (int chunk = 0; chunk < C_ / 32; ++chunk) {
        const int    c    = (chunk << 5) + lane;
        const size_t coff = (size_t)c * P_;
        float v;
        v  = w000 * feat[i000 + coff];
        v += w001 * feat[i001 + coff];
        v += w010 * feat[i010 + coff];
        v += w011 * feat[i011 + coff];
        v += w100 * feat[i100 + coff];
        v += w101 * feat[i101 + coff];
        v += w110 * feat[i110 + coff];
        v += w111 * feat[i111 + coff];
        buf[c * CELLS + s] = v;
      }
    }
  }
  __syncthreads();

  // Write out (N, C, 8, 7, 7): out[n][c][ft][s]; streamed once -> NT stores
  // keep the (mostly L2-resident) transposed feature tensor from eviction.
  const size_t outbase = ((size_t)n * C_) * OUTSTRIDE + (size_t)ft * CELLS;
  for (int j = tid; j < CELLS * C_; j += 256) {
    const int c = j / CELLS;
    const int s = j - c * CELLS;
    __builtin_nontemporal_store(buf[j],
                                &out[outbase + (size_t)c * OUTSTRIDE + s]);
  }
}

// ---------------------------------------------------------------------------
extern "C" void kernel_launch(void* const* d_in, const int* in_sizes, int n_in,
                              void* d_out, int out_size, void* d_ws,
                              size_t ws_size, hipStream_t stream) {
  (void)in_sizes; (void)n_in; (void)out_size;
  const float* feat = (const float*)d_in[0];
  const float* rois = (const float*)d_in[1];
  float* out = (float*)d_out;

  const size_t need = (size_t)B_ * C_ * P_ * sizeof(float);  // 205.5 MB
  if (d_ws != nullptr && ws_size >= need) {
    float* tf = (float*)d_ws;
    dim3 grid(P_ / 32, C_ / 32, B_);  // (1568, 8, 4)
    dim3 blk(32, 8, 1);
    transpose_cthw_to_thwc<<<grid, blk, 0, stream>>>(feat, tf);
    toi_pool<<<N_ * FR, 256, 0, stream>>>(tf, rois, out, 1);
  } else {
    // Workspace too small: correct (slower) direct-gather fallback
    toi_pool<<<N_ * FR, 256, 0, stream>>>(feat, rois, out, 0);
  }
}